// DeChunkLayer_47880295415816
// MI455X (gfx1250) — compile-verified
//
#include <hip/hip_runtime.h>

// Problem constants (from reference setup_inputs)
#define BB 4
#define LL 8192
#define MM 4096
#define DD 1024
#define NC 256           // MM / 16 chunks
#define EPSV 1e-4f

typedef float v2f __attribute__((ext_vector_type(2)));
typedef float v8f __attribute__((ext_vector_type(8)));

// ---- bf16 helpers ---------------------------------------------------------
// Exact widen (bf16 -> f32 is lossless).
__device__ __forceinline__ float bf2f(unsigned short u) {
    union { unsigned int i; float f; } v;
    v.i = ((unsigned int)u) << 16;
    return v.f;
}
// Branchless round-to-nearest-even narrow. All values reaching this in the
// pipeline are finite (p clipped to [eps,1-eps] => dt>0; hidden ~ N(0,1)),
// so no NaN/Inf guard is needed -> no divergent EXEC branches in hot loops.
__device__ __forceinline__ unsigned short f2bf(float f) {
    union { float f; unsigned int i; } v;
    v.f = f;
    unsigned int x = v.i;
    x += 0x7fffu + ((x >> 16) & 1u);
    return (unsigned short)(x >> 16);
}
__device__ __forceinline__ float bf16_round(float f) { return bf2f(f2bf(f)); }

// ---------------------------------------------------------------------------
// Kernel A: per-batch cumsum of boundary_mask; scatter clipped p into sorted
// order (boundary tokens first, stable); compute plug_back_idx.
// grid = BB, block = 256
// ---------------------------------------------------------------------------
__global__ __launch_bounds__(256) void boundary_kernel(
    const int* __restrict__ bmask,    // [B,L] (0/1)
    const float* __restrict__ bprob,  // [B,L,2]
    int* __restrict__ idx,            // [B,L]
    float* __restrict__ p_sorted)     // [B,M]
{
    __shared__ int partial[256];
    const int b = blockIdx.x;
    const int t = threadIdx.x;
    const int seg = LL / 256;         // 32
    const int* bm = bmask + (size_t)b * LL;

    int s = 0;
    for (int i = 0; i < seg; ++i) s += (bm[t * seg + i] != 0);
    partial[t] = s;
    __syncthreads();
    // inclusive Hillis-Steele scan over 256 partials
    for (int off = 1; off < 256; off <<= 1) {
        int add = (t >= off) ? partial[t - off] : 0;
        __syncthreads();
        partial[t] += add;
        __syncthreads();
    }
    const int base = partial[t] - s;   // exclusive prefix for my segment
    const int n_b  = partial[255];     // total boundary tokens this batch

    const float* pp = bprob + (size_t)b * LL * 2;
    int cb = base;                     // inclusive count as we walk segment
    for (int i = 0; i < seg; ++i) {
        const int l = t * seg + i;
        const bool ism = (bm[l] != 0);
        int pos;
        if (ism) { cb += 1; pos = cb - 1; }
        else     { pos = n_b + (l - cb); }
        if (pos < MM) {
            float pv = pp[(size_t)l * 2 + 1];
            pv = fminf(fmaxf(pv, EPSV), 1.0f - EPSV);
            p_sorted[(size_t)b * MM + pos] = pv;
        }
        int ii = cb - 1;
        ii = (ii < 0) ? 0 : ((ii > MM - 1) ? MM - 1 : ii);
        idx[(size_t)b * LL + l] = ii;
    }
}

// ---------------------------------------------------------------------------
// Kernel B: per 16-row chunk build W'[16][16], cumA[16], rdt[16].
//   dt = bf16_round(log(1/(1-p)));  a = exp(-dt);  c = dt*p;  rdt = 1/dt
//   W'[i][j] = c_j * prod_{k=j+1..i} a_k  (j<=i),  cumA[i] = prod_{0..i} a_k
// grid = BB*NC, block = 32 (lanes >=16 idle)
// ---------------------------------------------------------------------------
__global__ __launch_bounds__(32) void chunkw_kernel(
    const float* __restrict__ p_sorted, // [B,M]
    float* __restrict__ Wp,             // [B,NC,16,16]
    float* __restrict__ cumA,           // [B,NC,16]
    float* __restrict__ rdtf)           // [B,M]  (1/dt, exact IEEE div once)
{
    __shared__ float a_sh[16], c_sh[16];
    const int blk = blockIdx.x;
    const int b = blk >> 8;            // / NC
    const int t = blk & (NC - 1);
    const int i = threadIdx.x;

    if (i < 16) {
        const int m = t * 16 + i;
        const float p = p_sorted[(size_t)b * MM + m];
        float dt = logf(1.0f / (1.0f - p));
        dt = bf16_round(dt);
        const float a = expf(-dt);
        a_sh[i] = a;
        c_sh[i] = dt * p;
        rdtf[(size_t)b * MM + m] = 1.0f / dt;
    }
    __syncthreads();
    if (i < 16) {
        float* Wrow = Wp + ((size_t)(b * NC + t) * 256) + i * 16;
        Wrow[i] = c_sh[i];
        for (int j = i + 1; j < 16; ++j) Wrow[j] = 0.0f;
        float prod = 1.0f;
        for (int j = i - 1; j >= 0; --j) {
            prod *= a_sh[j + 1];
            Wrow[j] = c_sh[j] * prod;
        }
        float cp = 1.0f;
        for (int k = 0; k <= i; ++k) cp *= a_sh[k];
        cumA[((size_t)(b * NC + t)) * 16 + i] = cp;
    }
}

// ---------------------------------------------------------------------------
// Kernel C: WMMA chunked scan. One wave per (batch, 16-col d-tile):
//   for each chunk t: H = W' @ bf16(X * rdt)  +  cumA (x) carry  (f32 WMMA)
// carry = row 15 of H, pulled via lane shuffle. States stored as bf16.
// grid = BB*64, block = 32
// ---------------------------------------------------------------------------
__global__ __launch_bounds__(32) void scan_wmma_kernel(
    const float* __restrict__ hidden,           // [B,M,D]
    const float* __restrict__ Wp,               // [B,NC,16,16]
    const float* __restrict__ cumA,             // [B,NC,16]
    const float* __restrict__ rdtf,             // [B,M]
    unsigned short* __restrict__ hout)          // [B,M,D] bf16 bits
{
    const int lane  = threadIdx.x;
    const int b     = blockIdx.x >> 6;          // / 64
    const int dtile = blockIdx.x & 63;
    const int d0    = dtile * 16;
    const int n     = lane & 15;
    const int hi    = lane >> 4;                // 0 | 1
    const int koff  = hi * 2;                   // K half per lane group

    const float* Wb  = Wp   + (size_t)b * NC * 256;
    const float* cAb = cumA + (size_t)b * NC * 16;
    const float* dtb = rdtf + (size_t)b * MM;
    const float* hb  = hidden + (size_t)b * MM * DD + d0;
    unsigned short* ob = hout + (size_t)b * MM * DD + d0;

    float carry = 0.0f;

    for (int t = 0; t < NC; ++t) {
        const int m0 = t * 16;
        const float* Wt = Wb + (size_t)t * 256;

        if (t + 1 < NC) {  // prefetch next chunk's X rows and W' tile
            __builtin_prefetch(hb + (size_t)(m0 + 16 + n) * DD, 0, 1);
            __builtin_prefetch(Wt + 256 + lane * 8, 0, 1);
        }

        // A operand (16x4 f32): lane row i=n, K pair {4k+koff, 4k+koff+1}
        v2f A[4];
#pragma unroll
        for (int k = 0; k < 4; ++k) {
            A[k][0] = Wt[n * 16 + 4 * k + koff + 0];
            A[k][1] = Wt[n * 16 + 4 * k + koff + 1];
        }
        // B operand (4x16 f32): lane col n, rows {4k+koff, 4k+koff+1}
        v2f Bv[4];
#pragma unroll
        for (int k = 0; k < 4; ++k) {
            const int r0 = 4 * k + koff;
            const float x0 = hb[(size_t)(m0 + r0 + 0) * DD + n] * dtb[m0 + r0 + 0];
            const float x1 = hb[(size_t)(m0 + r0 + 1) * DD + n] * dtb[m0 + r0 + 1];
            Bv[k][0] = bf16_round(x0);
            Bv[k][1] = bf16_round(x1);
        }
        // C init: rank-1 carry term  C[r] = cumA[r + hi*8] * carry[n]
        v8f C;
#pragma unroll
        for (int r = 0; r < 8; ++r)
            C[r] = cAb[(size_t)t * 16 + r + hi * 8] * carry;

        // H = W' @ X + C  via 4 chained f32 WMMAs (K=4 each)
#pragma unroll
        for (int k = 0; k < 4; ++k)
            C = __builtin_amdgcn_wmma_f32_16x16x4_f32(
                    false, A[k], false, Bv[k], (short)0, C, false, false);

        // store H as bf16 (branchless narrow -> plain b16 stores)
#pragma unroll
        for (int r = 0; r < 8; ++r)
            ob[(size_t)(m0 + r + hi * 8) * DD + n] = f2bf(C[r]);

        // carry = H[15][n] lives in lane 16+n of C[7]
        carry = __shfl(C[7], 16 + n, 32);
    }
}

// ---------------------------------------------------------------------------
// Kernel D: gather states back to token space, widen bf16 -> f32.
// grid = BB*LL, block = 256, 4 elems/thread (8B bf16 read, 16B f32 write)
// ---------------------------------------------------------------------------
__global__ __launch_bounds__(256) void gather_kernel(
    const unsigned short* __restrict__ hbf, // [B,M,D] bf16 bits
    const int* __restrict__ idx,            // [B,L]
    float* __restrict__ out)                // [B,L,D]
{
    const int bl = blockIdx.x;
    const int b  = bl / LL;
    const int m  = idx[bl];
    const unsigned short* src =
        hbf + ((size_t)b * MM + m) * DD + threadIdx.x * 4;
    const ushort4 raw = *(const ushort4*)src;
    float4 o;
    o.x = bf2f(raw.x);
    o.y = bf2f(raw.y);
    o.z = bf2f(raw.z);
    o.w = bf2f(raw.w);
    *(float4*)(out + (size_t)bl * DD + threadIdx.x * 4) = o;
}

// ---------------------------------------------------------------------------
extern "C" void kernel_launch(void* const* d_in, const int* in_sizes, int n_in,
                              void* d_out, int out_size, void* d_ws, size_t ws_size,
                              hipStream_t stream) {
    (void)in_sizes; (void)n_in; (void)out_size; (void)ws_size;

    const float* hidden = (const float*)d_in[0];   // [B,M,D] f32
    const float* bprob  = (const float*)d_in[1];   // [B,L,2] f32
    const int*   bmask  = (const int*)d_in[2];     // [B,L]   int
    // d_in[3] = mask, unused by the reference computation
    float* out = (float*)d_out;                    // [B,L,D] f32

    // workspace layout (bytes)
    char* ws = (char*)d_ws;
    int*            idx      = (int*)(ws + 0);                       // 128 KiB
    float*          p_sorted = (float*)(ws + 131072);                //  64 KiB
    float*          rdtf     = (float*)(ws + 196608);                //  64 KiB
    float*          cumA     = (float*)(ws + 262144);                //  64 KiB
    float*          Wp       = (float*)(ws + 327680);                //   1 MiB
    unsigned short* hbf      = (unsigned short*)(ws + 1376256);      //  32 MiB

    boundary_kernel<<<BB, 256, 0, stream>>>(bmask, bprob, idx, p_sorted);
    chunkw_kernel<<<BB * NC, 32, 0, stream>>>(p_sorted, Wp, cumA, rdtf);
    scan_wmma_kernel<<<BB * 64, 32, 0, stream>>>(hidden, Wp, cumA, rdtf, hbf);
    gather_kernel<<<BB * LL, 256, 0, stream>>>(hbf, idx, out);
}